// GATLayer_2430951489854
// MI455X (gfx1250) — compile-verified
//
#include <hip/hip_runtime.h>
#include <hip/hip_bf16.h>
#include <math.h>

typedef __attribute__((ext_vector_type(2))) float v2f;
typedef __attribute__((ext_vector_type(8))) float v8f;

#define IN_F   128
#define OUT_F  32
#define NHEAD  4
#define ALPHA  0.2f

// -------- float atomic max via monotone int-compare trick (always compiles) ----
__device__ inline void atomicMaxFloat(float* addr, float v) {
    int iv = __float_as_int(v);
    if (iv >= 0) {
        atomicMax((int*)addr, iv);                 // both ends non-neg or stored neg -> replaced
    } else {
        atomicMin((unsigned int*)addr, (unsigned int)iv); // neg floats: larger unsigned = more negative
    }
}

// -------- init: m = -inf, s = 0, out = 0 --------------------------------------
__global__ __launch_bounds__(256) void gat_init(float* __restrict__ m,
                                                float* __restrict__ s,
                                                float* __restrict__ out, int N) {
    long i = (long)blockIdx.x * blockDim.x + threadIdx.x;
    long n4 = (long)N * NHEAD;
    long nF = (long)N * (NHEAD * OUT_F);
    if (i < n4) { m[i] = -INFINITY; s[i] = 0.0f; }
    if (i < nF) { out[i] = 0.0f; }
}

// -------- h = x @ W via V_WMMA_F32_16X16X4_F32 --------------------------------
// One wave computes a 16-row x 128-col tile. Block = 256 threads = 8 waves = 128 rows.
__global__ __launch_bounds__(256) void gat_gemm(const float* __restrict__ x,
                                                const float* __restrict__ W,
                                                float* __restrict__ h, int N) {
    const int wave = threadIdx.x >> 5;
    const int lane = threadIdx.x & 31;
    const long rowTile = (long)blockIdx.x * 8 + wave;
    const long row0 = rowTile * 16;
    if (row0 >= N) return;                     // wave-uniform exit

    const int mrow  = lane & 15;               // M (for A) / N (for B) index within tile
    const int khalf = (lane >> 4) * 2;         // lanes 0-15 -> K+{0,1}; lanes 16-31 -> K+{2,3}

    v8f acc[8];
    #pragma unroll
    for (int t = 0; t < 8; ++t) acc[t] = v8f{};

    long r = row0 + mrow;
    if (r >= N) r = N - 1;                     // clamp tail reads; stores are guarded below
    const float* __restrict__ xrow = x + r * IN_F;

    for (int k = 0; k < IN_F; k += 4) {
        v2f a;
        a.x = xrow[k + khalf + 0];             // A 16x4 f32: V0 = K0|K2, V1 = K1|K3
        a.y = xrow[k + khalf + 1];
        #pragma unroll
        for (int t = 0; t < 8; ++t) {
            const int ncol = t * 16 + mrow;
            v2f b;
            b.x = W[(k + khalf + 0) * (OUT_F * NHEAD) + ncol];
            b.y = W[(k + khalf + 1) * (OUT_F * NHEAD) + ncol];
            // 8 args: (neg_a, A, neg_b, B, c_mod, C, reuse_a, reuse_b)
            acc[t] = __builtin_amdgcn_wmma_f32_16x16x4_f32(
                false, a, false, b, (short)0, acc[t], false, false);
        }
    }

    // C/D layout: VGPR g -> lanes 0-15: (M=g, N=lane); lanes 16-31: (M=g+8, N=lane-16)
    const int rowOff = (lane < 16) ? 0 : 8;
    #pragma unroll
    for (int t = 0; t < 8; ++t) {
        #pragma unroll
        for (int g = 0; g < 8; ++g) {
            long rr = row0 + g + rowOff;
            if (rr < N) {
                float v = acc[t][g];
                v = (v == v) ? v : 0.0f;       // nan_to_num
                h[rr * (OUT_F * NHEAD) + t * 16 + mrow] = v;
            }
        }
    }
}

// -------- al[n][j] = dot(a_l[j], h[n, j*32:(j+1)*32]); same for ar ------------
// One wave per node; lane covers one feature per head, shuffle-reduce within wave32.
__global__ __launch_bounds__(256) void gat_alar(const float* __restrict__ h,
                                                const float* __restrict__ a_l,
                                                const float* __restrict__ a_r,
                                                float* __restrict__ al,
                                                float* __restrict__ ar, int N) {
    const int wave = threadIdx.x >> 5;
    const int lane = threadIdx.x & 31;
    const long n = (long)blockIdx.x * 8 + wave;
    if (n >= N) return;
    const float* __restrict__ hn = h + n * (OUT_F * NHEAD);
    float sl[NHEAD], sr[NHEAD];
    #pragma unroll
    for (int j = 0; j < NHEAD; ++j) {
        float hv = hn[j * OUT_F + lane];
        sl[j] = a_l[j * OUT_F + lane] * hv;
        sr[j] = a_r[j * OUT_F + lane] * hv;
    }
    #pragma unroll
    for (int off = 16; off > 0; off >>= 1) {
        #pragma unroll
        for (int j = 0; j < NHEAD; ++j) {
            sl[j] += __shfl_xor(sl[j], off, 32);
            sr[j] += __shfl_xor(sr[j], off, 32);
        }
    }
    if (lane == 0) {
        #pragma unroll
        for (int j = 0; j < NHEAD; ++j) {
            al[n * NHEAD + j] = sl[j];
            ar[n * NHEAD + j] = sr[j];
        }
    }
}

// -------- per-edge raw score + segment max over destination (row) -------------
__global__ __launch_bounds__(256) void gat_edge_score(const int* __restrict__ row,
                                                      const int* __restrict__ col,
                                                      const float* __restrict__ al,
                                                      const float* __restrict__ ar,
                                                      float* __restrict__ e,
                                                      float* __restrict__ m, int E) {
    long i = (long)blockIdx.x * blockDim.x + threadIdx.x;
    if (i >= E) return;
    const int r = row[i], c = col[i];
    #pragma unroll
    for (int j = 0; j < NHEAD; ++j) {
        float v = al[(long)r * NHEAD + j] + ar[(long)c * NHEAD + j];
        v = (v > 0.0f) ? v : (ALPHA * v);      // leaky relu
        e[i * NHEAD + j] = v;
        atomicMaxFloat(&m[(long)r * NHEAD + j], v);
    }
}

// -------- e_exp = exp(e - m[row]); segment sum --------------------------------
__global__ __launch_bounds__(256) void gat_edge_exp(const int* __restrict__ row,
                                                    const float* __restrict__ m,
                                                    float* __restrict__ e,
                                                    float* __restrict__ s, int E) {
    long i = (long)blockIdx.x * blockDim.x + threadIdx.x;
    if (i >= E) return;
    const int r = row[i];
    #pragma unroll
    for (int j = 0; j < NHEAD; ++j) {
        float v = expf(e[i * NHEAD + j] - m[(long)r * NHEAD + j]);
        e[i * NHEAD + j] = v;
        atomicAdd(&s[(long)r * NHEAD + j], v);
    }
}

// -------- out[row] += att * h[col]; one wave per edge, lane = feature ---------
__global__ __launch_bounds__(256) void gat_aggregate(const int* __restrict__ row,
                                                     const int* __restrict__ col,
                                                     const float* __restrict__ e,
                                                     const float* __restrict__ s,
                                                     const float* __restrict__ h,
                                                     float* __restrict__ out, int E) {
    const int wave = threadIdx.x >> 5;
    const int lane = threadIdx.x & 31;
    const long i = (long)blockIdx.x * 8 + wave;
    if (i >= E) return;
    const int r = row[i], c = col[i];
    const float* __restrict__ hc = h + (long)c * (OUT_F * NHEAD);
    float* __restrict__ outr = out + (long)r * (OUT_F * NHEAD);
    #pragma unroll
    for (int j = 0; j < NHEAD; ++j) {
        float att = e[i * NHEAD + j] / s[(long)r * NHEAD + j];
        atomicAdd(&outr[j * OUT_F + lane], att * hc[j * OUT_F + lane]);
    }
}

extern "C" void kernel_launch(void* const* d_in, const int* in_sizes, int n_in,
                              void* d_out, int out_size, void* d_ws, size_t ws_size,
                              hipStream_t stream) {
    const float* x    = (const float*)d_in[0];
    const int*   edge = (const int*)d_in[1];
    const float* W    = (const float*)d_in[2];
    const float* a_l  = (const float*)d_in[3];
    const float* a_r  = (const float*)d_in[4];

    const int N = in_sizes[0] / IN_F;
    const int E = in_sizes[1] / 2;
    const int* rowp = edge;        // edge_index[0]
    const int* colp = edge + E;    // edge_index[1]

    // workspace layout (fp32): h[N*128] | al[N*4] | ar[N*4] | m[N*4] | s[N*4] | e[E*4]
    float* ws = (float*)d_ws;
    float* h  = ws;
    float* al = h  + (size_t)N * (OUT_F * NHEAD);
    float* ar = al + (size_t)N * NHEAD;
    float* m  = ar + (size_t)N * NHEAD;
    float* s  = m  + (size_t)N * NHEAD;
    float* e  = s  + (size_t)N * NHEAD;
    float* out = (float*)d_out;

    {   // init m/s/out
        long total = (long)N * (OUT_F * NHEAD);
        gat_init<<<(unsigned)((total + 255) / 256), 256, 0, stream>>>(m, s, out, N);
    }
    gat_gemm<<<(unsigned)((N + 127) / 128), 256, 0, stream>>>(x, W, h, N);
    gat_alar<<<(unsigned)((N + 7) / 8), 256, 0, stream>>>(h, a_l, a_r, al, ar, N);
    gat_edge_score<<<(unsigned)((E + 255) / 256), 256, 0, stream>>>(rowp, colp, al, ar, e, m, E);
    gat_edge_exp<<<(unsigned)((E + 255) / 256), 256, 0, stream>>>(rowp, m, e, s, E);
    gat_aggregate<<<(unsigned)((E + 7) / 8), 256, 0, stream>>>(rowp, colp, e, s, h, out, E);
}